// KinematicLayer_83176336654509
// MI455X (gfx1250) — compile-verified
//
#include <hip/hip_runtime.h>
#include <hip/hip_bf16.h>
#include <cstdint>

namespace {
constexpr int kNGP = 9;            // gauss points per element
constexpr int kNEN = 8;            // nodes per element
constexpr int kNB  = 4;            // batch
constexpr int kEPB = 32;           // elements per block
constexpr int kTPB = kEPB * kNGP;  // 288 threads = 9 wave32
}

typedef int v4i __attribute__((ext_vector_type(4)));

#if defined(__gfx1250__)
#define AS1 __attribute__((address_space(1)))
#define AS3 __attribute__((address_space(3)))

// cpol: gfx12-style CPol immediate, TH in [2:0]; TH_LOAD_NT = 1.
// shpdx is a read-once 288 MB stream -> non-temporal so it doesn't evict the
// L2-resident gather working set.
#if __has_builtin(__builtin_amdgcn_global_load_async_to_lds_b128)
// Signature (from clang diagnostic): (v4i AS1*, v4i AS3*, imm offset, imm cpol)
#define ASYNC_CP_B128(gp, lp, OFF)                                          \
  __builtin_amdgcn_global_load_async_to_lds_b128(                           \
      (AS1 v4i*)(void*)(gp), (AS3 v4i*)(void*)(lp), (OFF), 1)
#else
#define ASYNC_CP_B128(gp, lp, OFF)                                          \
  asm volatile("global_load_async_to_lds_b128 %0, %1, off offset:" #OFF     \
               " th:TH_LOAD_NT"                                             \
               :: "v"((unsigned)(uintptr_t)(AS3 void*)(lp)),                \
                  "v"((const void*)(gp))                                    \
               : "memory")
#endif

__device__ __forceinline__ void wait_async0() {
#if __has_builtin(__builtin_amdgcn_s_wait_asynccnt)
  __builtin_amdgcn_s_wait_asynccnt(0);
#else
  asm volatile("s_wait_asynccnt 0" ::: "memory");
#endif
}
#else
// Host pass / non-gfx1250: semantically equivalent plain copy so the host
// compilation of the kernel body stays clean.
#define ASYNC_CP_B128(gp, lp, OFF)                                          \
  do { __builtin_memcpy((char*)(lp) + (OFF), (const char*)(gp) + (OFF), 16); } while (0)
__device__ __forceinline__ void wait_async0() {}
#endif

// out[b, e*9+g, c] : c in {e_xx, e_yy, e_xy}
__global__ __launch_bounds__(kTPB) void kinematic_strain_kernel(
    const float* __restrict__ inputs,      // [B, n_nodes, 2]
    const float* __restrict__ shpdx,       // [n_elem, 9, 8, 2]
    const int*   __restrict__ elem_nodes,  // [n_elem, 8]
    float* __restrict__ out,               // [B, n_elem*9, 3]
    int n_elem, int n_nodes)
{
  __shared__ __align__(16) float lds_shp [kTPB * 16];              // 18 KB
  __shared__ __align__(16) float lds_disp[kEPB * kNB * kNEN * 2];  //  8 KB

  const int t      = threadIdx.x;
  const int e_base = blockIdx.x * kEPB;

  // ---- Phase 0: CDNA5 async DMA of this block's shpdx tile into LDS ------
  // flat (element, gauss point) index; consecutive threads -> consecutive
  // 64B chunks -> fully coalesced streaming read, tracked by ASYNCcnt.
  const int eg = e_base * kNGP + t;
  if (eg < n_elem * kNGP) {
    const float* g = shpdx + (size_t)eg * 16;
    float*       l = &lds_shp[t * 16];
    ASYNC_CP_B128(g, l, 0);
    ASYNC_CP_B128(g, l, 16);
    ASYNC_CP_B128(g, l, 32);
    ASYNC_CP_B128(g, l, 48);
  }

  // ---- Phase 1: gather nodal displacements once per element (not per GP) --
  if (t < kEPB * kNEN) {                 // 256 gather threads
    const int el = t >> 3;               // element within block: 0..31
    const int k  = t & 7;                // node slot: 0..7
    const int e  = e_base + el;
    if (e < n_elem) {
      // connectivity is read exactly once, coalesced -> non-temporal
      const int node =
          __builtin_nontemporal_load(elem_nodes + (size_t)e * kNEN + k);
#pragma unroll
      for (int b = 0; b < kNB; ++b) {
        const float2 u =
            *(const float2*)(inputs + ((size_t)b * n_nodes + (size_t)node) * 2);
        *(float2*)&lds_disp[(((el * kNB) + b) * kNEN + k) * 2] = u;
      }
    }
  }
  __syncthreads();   // publish lds_disp to all 9 waves
  wait_async0();     // this wave's own shpdx slots are now in LDS

  // ---- Phase 2: per-(element,GP) strain ----------------------------------
  const int el = t / kNGP;               // element within block
  const int e  = e_base + el;
  if (e >= n_elem) return;

  float s[16];                           // shpdx[e,g,k,l]: s[2k]=d/dx, s[2k+1]=d/dy
  {
    const float* S = &lds_shp[t * 16];
#pragma unroll
    for (int i = 0; i < 4; ++i)
      *(float4*)&s[i * 4] = *(const float4*)(S + i * 4);
  }

  const size_t bstride = (size_t)n_elem * kNGP * 3;
  const size_t obase   = (size_t)eg * 3;

#pragma unroll
  for (int b = 0; b < kNB; ++b) {
    float u[16];                         // disp[k,n]: u[2k]=ux, u[2k+1]=uy
    const float* U = &lds_disp[((el * kNB) + b) * kNEN * 2];
#pragma unroll
    for (int i = 0; i < 4; ++i)
      *(float4*)&u[i * 4] = *(const float4*)(U + i * 4);

    float exx = 0.f, eyy = 0.f, exy2 = 0.f;
#pragma unroll
    for (int k = 0; k < kNEN; ++k) {
      const float sx = s[2 * k], sy = s[2 * k + 1];
      const float ux = u[2 * k], uy = u[2 * k + 1];
      exx  = __builtin_fmaf(sx, ux, exx);    // dd[0,0]
      eyy  = __builtin_fmaf(sy, uy, eyy);    // dd[1,1]
      exy2 = __builtin_fmaf(sy, ux, exy2);   // dd[0,1]
      exy2 = __builtin_fmaf(sx, uy, exy2);   // + dd[1,0]
    }

    float* o = out + (size_t)b * bstride + obase;
    // write-once 216 MB stream -> non-temporal stores (merge to b96)
    __builtin_nontemporal_store(exx,          o + 0);
    __builtin_nontemporal_store(eyy,          o + 1);
    __builtin_nontemporal_store(0.5f * exy2,  o + 2);
  }
}

extern "C" void kernel_launch(void* const* d_in, const int* in_sizes, int n_in,
                              void* d_out, int out_size, void* d_ws, size_t ws_size,
                              hipStream_t stream) {
  const float* inputs     = (const float*)d_in[0];
  const float* shpdx      = (const float*)d_in[1];
  const int*   elem_nodes = (const int*)d_in[2];
  float*       out        = (float*)d_out;

  const int n_elem  = in_sizes[2] / kNEN;        // elem_nodes is [nE, 8]
  const int n_nodes = in_sizes[0] / (kNB * 2);   // inputs is [4, nN, 2]

  const int blocks = (n_elem + kEPB - 1) / kEPB; // 15625 for nE=500K

  kinematic_strain_kernel<<<blocks, kTPB, 0, stream>>>(
      inputs, shpdx, elem_nodes, out, n_elem, n_nodes);

  (void)n_in; (void)out_size; (void)d_ws; (void)ws_size;
}